// CoEvolutionLoss_70497593196923
// MI455X (gfx1250) — compile-verified
//
#include <hip/hip_runtime.h>
#include <hip/hip_bf16.h>
#include <stdint.h>

typedef __attribute__((ext_vector_type(2))) float v2f;
typedef __attribute__((ext_vector_type(8))) float v8f;

#define N_PTS 8192     // 16 * 512
#define DIM   32
#define TILES 512      // 8192 / 16

// Per-codon biosynthetic family id, codon order = Python dict insertion order.
// glutamate=0, aspartate=1 (K overwritten into aspartate), serine=2,
// pyruvate=3, aromatic=4, histidine=5, stop=-1.
__constant__ int FID_TBL[64] = {
    4, 4, 3, 3,   3, 3, 3, 3,   1, 1, 1, 1,   3, 3, 3, 3,
    2, 2, 2, 2,   0, 0, 0, 0,   1, 1, 1, 1,   3, 3, 3, 3,
    4, 4,-1,-1,   5, 5, 0, 0,   1, 1, 1, 1,   1, 1, 0, 0,
    2, 2,-1, 4,   0, 0, 0, 0,   2, 2, 0, 0,   2, 2, 2, 2
};

__global__ void coevo_prep(const float* __restrict__ x,
                           const long long* __restrict__ idx,
                           float* __restrict__ sq,
                           int* __restrict__ fam,
                           float* __restrict__ acc,
                           unsigned* __restrict__ cnt) {
    int m = blockIdx.x * blockDim.x + threadIdx.x;
    if (m == 0) { acc[0] = 0.f; acc[1] = 0.f; cnt[0] = 0u; }
    if (m < N_PTS) {
        const float* row = x + m * DIM;
        float s = 0.f;
#pragma unroll
        for (int k = 0; k < DIM; ++k) s += row[k] * row[k];
        sq[m] = s;
        fam[m] = FID_TBL[((int)idx[m]) & 63];
    }
}

// One wave32 per 16x16 tile of the 8192x8192 distance matrix.
__global__ void __launch_bounds__(256) coevo_pair(const float* __restrict__ x,
                                                  const float* __restrict__ sq,
                                                  const int* __restrict__ fam,
                                                  float* __restrict__ acc,
                                                  unsigned* __restrict__ cnt) {
    const int lane = threadIdx.x & 31;
    const int wave = threadIdx.x >> 5;
    const unsigned tile = blockIdx.x * 8u + wave;   // 0 .. 512*512-1
    const int ti = tile >> 9;                       // tile / 512
    const int tj = tile & 511;                      // tile % 512
    const int ib = ti << 4;
    const int jb = tj << 4;
    const int half = lane >> 4;                     // 0: lanes 0-15, 1: lanes 16-31
    const int l16  = lane & 15;

    // A (16x4 f32) layout: lanes 0-15 hold M=0..15; VGPR0 = K=0 (low half)/K=2
    // (high half), VGPR1 = K=1/K=3.  B (4x16) mirrors this for X*X^T.
    const float* rowA = x + (ib + l16) * DIM;
    const float* rowB = x + (jb + l16) * DIM;

    v8f c = {};
#pragma unroll
    for (int kk = 0; kk < 8; ++kk) {                // K = 32 = 8 * 4
        const int k0 = kk * 4 + half * 2;           // 8-byte aligned
        v2f a = *(const v2f*)(rowA + k0);
        v2f b = *(const v2f*)(rowB + k0);
        c = __builtin_amdgcn_wmma_f32_16x16x4_f32(
                /*neg_a=*/false, a, /*neg_b=*/false, b,
                /*c_mod=*/(short)0, c, /*reuse_a=*/false, /*reuse_b=*/false);
    }

    // C layout: VGPR r -> M = r + 8*half, lane -> N = l16.
    const float    sqn = sq[jb + l16];
    const int      fn  = fam[jb + l16];
    float    s_same = 0.f, s_diff = 0.f;
    unsigned c_same = 0u;
#pragma unroll
    for (int r = 0; r < 8; ++r) {
        const int   m   = ib + r + half * 8;
        const float d2  = sq[m] + sqn - 2.0f * c[r];
        const float dist = d2 > 0.f ? __builtin_sqrtf(d2) : 0.f;
        const bool  same = (fam[m] == fn) && (fn >= 0);
        s_same += same ? dist : 0.f;
        s_diff += same ? 0.f : dist;
        c_same += same ? 1u : 0u;
    }

    // wave32 butterfly reduction
#pragma unroll
    for (int off = 16; off > 0; off >>= 1) {
        s_same += __shfl_xor(s_same, off, 32);
        s_diff += __shfl_xor(s_diff, off, 32);
        c_same += __shfl_xor(c_same, off, 32);
    }
    if (lane == 0) {
        atomicAdd(&acc[0], s_same);
        atomicAdd(&acc[1], s_diff);
        atomicAdd(cnt, c_same);
    }
}

__global__ void coevo_final(const float* __restrict__ acc,
                            const unsigned* __restrict__ cnt,
                            float* __restrict__ out) {
    const float cs    = (float)cnt[0];
    const float total = (float)N_PTS * (float)N_PTS;
    const float cd    = total - cs;
    const float same_sum = acc[0] / (cs + 1e-10f);
    const float diff_sum = acc[1] / (cd + 1e-10f);
    const float loss = same_sum - 0.5f * diff_sum + 1.0f;
    out[0] = loss > 0.f ? loss : 0.f;
}

extern "C" void kernel_launch(void* const* d_in, const int* in_sizes, int n_in,
                              void* d_out, int out_size, void* d_ws, size_t ws_size,
                              hipStream_t stream) {
    const float*     x   = (const float*)d_in[0];        // (16,512,32) f32
    const long long* idx = (const long long*)d_in[1];    // (16,512) i64

    float*    sq  = (float*)d_ws;                                    // 8192 f32
    int*      fam = (int*)((char*)d_ws + N_PTS * sizeof(float));     // 8192 i32
    float*    acc = (float*)((char*)d_ws + 2u * N_PTS * sizeof(float)); // 2 f32
    unsigned* cnt = (unsigned*)(acc + 2);                            // 1 u32
    float*    out = (float*)d_out;

    coevo_prep<<<(N_PTS + 255) / 256, 256, 0, stream>>>(x, idx, sq, fam, acc, cnt);

    const int nblocks = (TILES * TILES) / 8;   // 8 waves (tiles) per 256-thread block
    coevo_pair<<<nblocks, 256, 0, stream>>>(x, sq, fam, acc, cnt);

    coevo_final<<<1, 1, 0, stream>>>(acc, cnt, out);
}